// _PeriodicConvOp_64596308131980
// MI455X (gfx1250) — compile-verified
//
#include <hip/hip_runtime.h>
#include <hip/hip_bf16.h>

typedef __attribute__((ext_vector_type(2))) float v2f;
typedef __attribute__((ext_vector_type(8))) float v8f;

#define C_IN   4
#define C_OUTC 8
#define HH     512
#define WW     512
#define HP     256
#define WP     256

// ---------------------------------------------------------------------------
// Prep kernel: build the zero-embedded GEMM weights W_emb[32 x 64], already
// swizzled into the per-lane A-fragment layout of V_WMMA_F32_16X16X4_F32:
//   ws[((mt*16 + k)*32 + lane)*2 + q]  =  A(mt,k) VGPR q for `lane`
// A 16x4 layout: lanes 0-15 hold K=0,1 in v0,v1; lanes 16-31 hold K=2,3.
// Rows m = c*4 + i*2 + j (== g*8 + co of the raw weight); K index
// kk = ch*16 + dy*4 + dx over the 4x4 stride-2 input window.
// ---------------------------------------------------------------------------
__global__ __launch_bounds__(256)
void build_wemb(const float* __restrict__ wgt, float* __restrict__ wemb) {
  const int t = blockIdx.x * blockDim.x + threadIdx.x;   // [0, 2048)
  if (t >= 2048) return;
  const int q    = t & 1;
  const int lane = (t >> 1) & 31;
  const int k    = (t >> 6) & 15;
  const int mt   = t >> 10;
  const int h = lane >> 4, p = lane & 15;
  const int row = 16 * mt + p;
  const int g   = row >> 3;
  const int yl  = g >> 1, xl = g & 1;
  const int kk  = 4 * k + 2 * h + q;
  const int ch  = kk >> 4;
  const int dy  = (kk >> 2) & 3;
  const int dx  = kk & 3;
  const int ky  = dy - yl, kx = dx - xl;
  float wv = 0.f;
  if (ky >= 0 && ky < 3 && kx >= 0 && kx < 3)
    wv = wgt[row * 36 + ch * 9 + ky * 3 + kx];
  wemb[t] = wv;
}

// ---------------------------------------------------------------------------
// Main kernel: one wave computes D[32 x 16] = W_emb[32 x 64] @ Patch[64 x 16]
// per N-tile, 4 N-tiles per wave. Patches are read straight from global (x
// fits in L2 many times over; ~4x read reuse is absorbed by WGP$/L2).
// ---------------------------------------------------------------------------
__global__ __launch_bounds__(128)
void periodic_conv_wmma_f32(const float* __restrict__ x,
                            const float* __restrict__ wemb,
                            float* __restrict__ out) {
  const int lane = threadIdx.x & 31;
  const int h    = lane >> 4;   // lane half: K pair (A) / dx pair (B)
  const int p    = lane & 15;   // M row (A) / pixel column (B,D)

  const int wid = blockIdx.x * (blockDim.x >> 5) + (threadIdx.x >> 5);
  const int ws  = wid & 3;                // quarter of the tile-row
  const int hp  = (wid >> 2) & (HP - 1);  // tile row
  const int b   = wid >> 10;              // batch

  // ---- Load pre-swizzled A fragments: 32 coalesced b64 loads -------------
  v2f afrag[2][16];
  const float2* we = (const float2*)wemb;
#pragma unroll
  for (int mt = 0; mt < 2; ++mt) {
#pragma unroll
    for (int k = 0; k < 16; ++k) {
      const float2 f = we[(mt * 16 + k) * 32 + lane];
      afrag[mt][k] = (v2f){f.x, f.y};
    }
  }

  const float* xb = x   + (size_t)b * C_IN   * HH * WW;
  float*       ob = out + (size_t)b * C_OUTC * HH * WW;

  for (int t = 0; t < 4; ++t) {
    const int wp0  = ws * 64 + t * 16;
    // B 4x16 f32 layout (mirror of A): per lane two adjacent x columns.
    const int col0 = 2 * (wp0 + p) + 2 * h - 1;   // dx = 2h
    const int col1 = col0 + 1;                    // dx = 2h+1
    const bool f0  = (col0 >= 0);
    const bool f1  = (col1 < WW);
    const int cc0  = f0 ? col0 : 0;
    const int cc1  = f1 ? col1 : (WW - 1);

    // ---- Batch all 16 K-steps' B fragments (loads overlap in flight) ----
    v2f bv[16];
#pragma unroll
    for (int k = 0; k < 16; ++k) {
      const int ch = k >> 2;
      const int dy = k & 3;
      const int y  = 2 * hp + dy - 1;
      const bool yok = (y >= 0) && (y < HH);
      const int  yc  = yok ? y : 0;
      const float* rp = xb + ((size_t)ch * HH + yc) * WW;
      float b0 = rp[cc0];
      float b1 = rp[cc1];
      b0 = (yok && f0) ? b0 : 0.f;
      b1 = (yok && f1) ? b1 : 0.f;
      bv[k] = (v2f){b0, b1};
    }

    // ---- WMMA chain: D = sum_k A_k * B_k ---------------------------------
    v8f acc0 = {};
    v8f acc1 = {};
#pragma unroll
    for (int k = 0; k < 16; ++k) {
      acc0 = __builtin_amdgcn_wmma_f32_16x16x4_f32(
          false, afrag[0][k], false, bv[k], (short)0, acc0, false, false);
      acc1 = __builtin_amdgcn_wmma_f32_16x16x4_f32(
          false, afrag[1][k], false, bv[k], (short)0, acc1, false, false);
    }

    // ---- Store D: rows (m, m+1) are the j=0/1 pair -> one b64 per lane.
    // D 16x16: VGPR v holds row v (lanes 0-15) and row v+8 (lanes 16-31).
#pragma unroll
    for (int mt = 0; mt < 2; ++mt) {
#pragma unroll
      for (int v = 0; v < 8; v += 2) {
        const float e0 = mt ? acc1[v]     : acc0[v];
        const float e1 = mt ? acc1[v + 1] : acc0[v + 1];
        const int c  = 4 * mt + (v >> 2) + 2 * h;  // output channel
        const int i  = (v >> 1) & 1;               // sub-row
        float* dst = ob + (((size_t)c * HH + (2 * hp + i)) * WW
                           + 2 * (wp0 + p));
        const v2f val = (v2f){e0, e1};
        __builtin_nontemporal_store(val, (v2f*)dst);
      }
    }
  }
}

extern "C" void kernel_launch(void* const* d_in, const int* in_sizes, int n_in,
                              void* d_out, int out_size, void* d_ws, size_t ws_size,
                              hipStream_t stream) {
  const float* x   = (const float*)d_in[0];
  const float* wgt = (const float*)d_in[1];
  float* out  = (float*)d_out;
  float* wemb = (float*)d_ws;   // 2048 floats (8 KB) of pre-swizzled weights

  hipLaunchKernelGGL(build_wemb, dim3(8), dim3(256), 0, stream, wgt, wemb);

  // 8192 waves = B(8) * HP(256) * 4 quarter-rows; 4 waves per 128-thread block.
  hipLaunchKernelGGL(periodic_conv_wmma_f32, dim3(2048), dim3(128), 0, stream,
                     x, wemb, out);
}